// ExactTopKAttention_73443940761951
// MI455X (gfx1250) — compile-verified
//
#include <hip/hip_runtime.h>
#include <math.h>

// Problem constants (match reference)
#define N_    2
#define L_    2048
#define S_    2048
#define H_    8
#define E_    64
#define D_    64
#define TOPK_ 32

typedef __attribute__((ext_vector_type(2))) float v2f;
typedef __attribute__((ext_vector_type(8))) float v8f;

#define NEG_INF (-INFINITY)

// One block: 256 threads (8 waves), handles one (n, h, 16-query tile).
//   Phase 1: scores[16][2048] = Q*K^T via v_wmma_f32_16x16x4_f32, key-length masked, in LDS
//   Phase 2: exact top-32 per row (iterative wave argmax, winner-only rescan)
//   Phase 3: softmax over selected 32 + gathered A*V
__global__ __launch_bounds__(256) void topk_attn_kernel(
    const float* __restrict__ Q,   // [N, L, H, E]
    const float* __restrict__ K,   // [N, S, H, E]
    const float* __restrict__ V,   // [N, S, H, D]
    const int*   __restrict__ KL,  // [N]
    float*       __restrict__ Out) // [N, L, H, D]
{
  extern __shared__ char smem_raw[];
  float* qs     = (float*)smem_raw;            // [16][E_]      4 KB
  float* scores = qs + 16 * E_;                // [16][S_]    128 KB
  float* topv   = scores + 16 * S_;            // [16][TOPK_]   2 KB
  float* topw   = topv + 16 * TOPK_;           // [16][TOPK_]   2 KB
  int*   topi   = (int*)(topw + 16 * TOPK_);   // [16][TOPK_]   2 KB
  float* denom  = (float*)(topi + 16 * TOPK_); // [16]

  const int q0   = blockIdx.x * 16;
  const int h    = blockIdx.y;
  const int n    = blockIdx.z;
  const int tid  = threadIdx.x;
  const int lane = tid & 31;
  const int wave = tid >> 5;
  const int keylen = KL[n];

  // ---- stage the 16x64 Q tile into LDS (coalesced) ----
  for (int i = tid; i < 16 * E_; i += 256) {
    const int m = i >> 6, e = i & 63;
    qs[i] = Q[(((size_t)n * L_ + q0 + m) * H_ + h) * E_ + e];
  }
  __syncthreads();

  // ---- Phase 1: QK^T scores via WMMA f32 16x16x4 ----
  // A (16x4 f32): lanes 0-15 hold M=0-15 with K={k0,k0+1}; lanes 16-31 K={k0+2,k0+3}
  // B (4x16 f32): lanes 0-15 hold N=0-15 with K={k0,k0+1}; lanes 16-31 K={k0+2,k0+3}
  // C/D (16x16 f32): VGPR r, lane l -> M = r + 8*(l>=16), N = l&15
  const int col = lane & 15;
  const int kb  = (lane >> 4) * 2;

  v2f a[16];
  #pragma unroll
  for (int kk = 0; kk < 16; ++kk) {
    a[kk] = *(const v2f*)(qs + col * E_ + kk * 4 + kb);
  }

  const float* kbase = K + (((size_t)n * S_) * H_ + h) * E_;
  for (int t = 0; t < 16; ++t) {
    const int s0 = (wave * 16 + t) * 16;           // wave w covers s in [w*256, w*256+256)
    const int s  = s0 + col;
    v8f acc = {0.f, 0.f, 0.f, 0.f, 0.f, 0.f, 0.f, 0.f};
    const float* kp = kbase + (size_t)s * (H_ * E_) + kb;
    #pragma unroll
    for (int kk = 0; kk < 16; ++kk) {
      v2f b = *(const v2f*)(kp + kk * 4);
      acc = __builtin_amdgcn_wmma_f32_16x16x4_f32(
          /*neg_a=*/false, a[kk], /*neg_b=*/false, b,
          /*c_mod=*/(short)0, acc, /*reuse_a=*/false, /*reuse_b=*/false);
    }
    const bool valid = (s < keylen);
    #pragma unroll
    for (int r = 0; r < 8; ++r) {
      const int m = r + ((lane >> 4) << 3);
      scores[m * S_ + s] = valid ? acc[r] : NEG_INF;
    }
  }
  __syncthreads();

  // ---- Phase 2: exact top-32 per row (one wave per row, 2 rows per wave) ----
  for (int rr = 0; rr < 2; ++rr) {
    const int row = wave + rr * 8;
    float* srow = scores + row * S_;

    // per-lane cached (max, argmax) over its 64 strided elements
    float lv = NEG_INF; int li = lane;
    for (int t2 = lane; t2 < S_; t2 += 32) {
      const float v = srow[t2];
      if (v > lv) { lv = v; li = t2; }
    }

    for (int j = 0; j < TOPK_; ++j) {
      float gv = lv; int gi = li;
      #pragma unroll
      for (int off = 16; off > 0; off >>= 1) {
        const float ov = __shfl_xor(gv, off, 32);
        const int   oi = __shfl_xor(gi, off, 32);
        if (ov > gv || (ov == gv && oi < gi)) { gv = ov; gi = oi; }
      }
      if (lane == 0) { topv[row * TOPK_ + j] = gv; topi[row * TOPK_ + j] = gi; }
      // winner lane (elements strided by 32 -> owner = gi & 31) invalidates + rescans
      if ((gi & 31) == lane) {
        srow[gi] = NEG_INF;
        lv = NEG_INF; li = lane;
        for (int t2 = lane; t2 < S_; t2 += 32) {
          const float v = srow[t2];
          if (v > lv) { lv = v; li = t2; }
        }
      }
    }
  }
  __syncthreads();

  // ---- Phase 2.5: softmax weights over the 32 selected scores ----
  const float temp = 0.125f; // 1/sqrt(E=64)
  for (int rr = 0; rr < 2; ++rr) {
    const int row = wave + rr * 8;
    const float maxv = topv[row * TOPK_];             // first selected == row max
    const float w = __expf(temp * (topv[row * TOPK_ + lane] - maxv));
    topw[row * TOPK_ + lane] = w;
    float ssum = w;
    #pragma unroll
    for (int off = 16; off > 0; off >>= 1) ssum += __shfl_xor(ssum, off, 32);
    if (lane == 0) denom[row] = ssum;
  }
  __syncthreads();

  // ---- Phase 3: gathered A*V (lanes <-> dims, broadcast weights) ----
  const float* vbase = V + (((size_t)n * S_) * H_ + h) * D_;
  for (int rr = 0; rr < 2; ++rr) {
    const int row = wave + rr * 8;
    float acc0 = 0.f, acc1 = 0.f;
    #pragma unroll 4
    for (int j = 0; j < TOPK_; ++j) {
      const float w  = topw[row * TOPK_ + j];
      const int  idx = topi[row * TOPK_ + j];
      const float* vr = vbase + (size_t)idx * (H_ * D_);
      acc0 += w * vr[lane];
      acc1 += w * vr[lane + 32];
    }
    const float inv = 1.f / denom[row];
    const size_t ob = (((size_t)n * L_ + q0 + row) * H_ + h) * D_;
    Out[ob + lane]      = acc0 * inv;
    Out[ob + lane + 32] = acc1 * inv;
  }
}

extern "C" void kernel_launch(void* const* d_in, const int* in_sizes, int n_in,
                              void* d_out, int out_size, void* d_ws, size_t ws_size,
                              hipStream_t stream) {
  const float* Q  = (const float*)d_in[0];
  const float* K  = (const float*)d_in[1];
  const float* V  = (const float*)d_in[2];
  const int*   KL = (const int*)d_in[3];
  float* Out = (float*)d_out;

  dim3 grid(L_ / 16, H_, N_);
  const size_t lds_bytes =
      (size_t)(16 * E_ + 16 * S_ + 3 * 16 * TOPK_ + 16) * sizeof(float);
  topk_attn_kernel<<<grid, dim3(256), lds_bytes, stream>>>(Q, K, V, KL, Out);
}